// Embeddings_46591805227295
// MI455X (gfx1250) — compile-verified
//
#include <hip/hip_runtime.h>
#include <hip/hip_bf16.h>

typedef __attribute__((ext_vector_type(2))) float v2f;
typedef __attribute__((ext_vector_type(8))) float v8f;

// out[row, d] = relu( x[row,0]*W[d,0] + x[row,1]*W[d,1] + bias[d] )
// rows = B*N = 512000, D_MODEL = 256.
// One block (256 thr = 8 waves) per 16-row tile; wave w covers d-tiles 2w, 2w+1.
__global__ void __launch_bounds__(256)
embed_linear_relu_wmma(const float* __restrict__ x,
                       const float* __restrict__ W,
                       const float* __restrict__ bias,
                       float* __restrict__ out,
                       int rows) {
    const int lane  = threadIdx.x & 31;
    const int wave  = threadIdx.x >> 5;          // 0..7
    const int laneM = lane & 15;
    const bool lo   = lane < 16;                 // lanes 0-15 carry K=0,1; 16-31 carry K=2,3 (zero pad)

    const int row0 = blockIdx.x * 16;
    if (row0 + 16 > rows) return;                // shapes divide exactly (512000 % 16 == 0)

    // A fragment: 16x4 f32. Lane 0-15: VGPR0=K0, VGPR1=K1 for row M=lane.
    // Lanes 16-31 would be K=2,3 -> zero (K padded 2->4).
    v2f a;
    if (lo) {
        const float* xp = x + (size_t)(row0 + laneM) * 2;
        a.x = xp[0];
        a.y = xp[1];
    } else {
        a.x = 0.0f; a.y = 0.0f;
    }

    #pragma unroll
    for (int t = 0; t < 2; ++t) {
        const int d0 = (wave * 2 + t) * 16;      // 0..240, covers all 256 channels

        // B fragment: 4x16 f32, B[k][n] = W[d0+n][k]. Lanes 0-15: VGPR0=K0 row,
        // VGPR1=K1 row, N=lane. Lanes 16-31 (K=2,3) -> zero.
        v2f bm;
        if (lo) {
            const float* wp = W + (size_t)(d0 + laneM) * 2;
            bm.x = wp[0];
            bm.y = wp[1];
        } else {
            bm.x = 0.0f; bm.y = 0.0f;
        }

        v8f c = {};
        // (neg_a, A, neg_b, B, c_mod, C, reuse_a, reuse_b)
        c = __builtin_amdgcn_wmma_f32_16x16x4_f32(
                false, a, false, bm, (short)0, c, false, false);

        // C/D layout: VGPR r -> row (row0 + r + (lane<16 ? 0 : 8)), col d0 + lane%16.
        const float bv = bias[d0 + laneM];
        float* op = out + (size_t)(row0 + (lo ? 0 : 8)) * 256 + d0 + laneM;
        #pragma unroll
        for (int r = 0; r < 8; ++r) {
            float v = c[r] + bv;
            op[(size_t)r * 256] = fmaxf(v, 0.0f);
        }
    }
}

extern "C" void kernel_launch(void* const* d_in, const int* in_sizes, int n_in,
                              void* d_out, int out_size, void* d_ws, size_t ws_size,
                              hipStream_t stream) {
    const float* x    = (const float*)d_in[0];   // [B, N, 2] f32
    const float* W    = (const float*)d_in[1];   // [256, 2]  f32
    const float* bias = (const float*)d_in[2];   // [256]     f32
    float*       out  = (float*)d_out;           // [B, N, 256] f32

    const int rows     = in_sizes[0] / 2;        // B*N = 512000
    const int rowTiles = rows / 16;              // 32000 blocks

    dim3 grid(rowTiles);
    dim3 block(256);
    hipLaunchKernelGGL(embed_linear_relu_wmma, grid, block, 0, stream,
                       x, W, bias, out, rows);
}